// Sort_Latent_Layer_3917010174779
// MI455X (gfx1250) — compile-verified
//
#include <hip/hip_runtime.h>
#include <stdint.h>

// Sort_Latent_Layer: per row of 8192 f32, stable-argsort 512 packets (16 f32
// each) by the packet's first element, then gather packets in sorted order.
//
// Design for MI455X (gfx1250):
//  - 1 workgroup (512 threads = 16 wave32) per row; 4096 rows.
//  - Stage the full 32KB row into LDS with CDNA5 async global->LDS loads
//    (global_load_async_to_lds_b128, ASYNCcnt-tracked), th:TH_LOAD_NT since
//    every input byte is read exactly once (256MB streaming vs 192MB L2).
//  - Unique 64-bit composite keys (IEEE-total-order bits << 32 | idx) make a
//    plain comparison sort reproduce jnp.argsort's *stable* order even when
//    f32 keys collide (expected in ~16 of 4096 rows).
//  - Bitonic sort (45 stages) on 512 composites in LDS: ~11.5K compare-
//    exchanges per row, far below the ~11us HBM roofline (256MB @ 23.3TB/s).
//  - Write-out in chunk-linear order: gathered reads come from LDS
//    (ds_load_b128), global stores are coalesced non-temporal b128.

typedef float v4f __attribute__((ext_vector_type(4)));  // native 16B vector

constexpr int P   = 16;        // floats per packet
constexpr int NPK = 512;       // packets per row
constexpr int D   = NPK * P;   // 8192 floats per row
constexpr int T   = 512;       // threads per block (16 waves)

__global__ __launch_bounds__(T)
void sort_latent_kernel(const float* __restrict__ z, float* __restrict__ out) {
    __shared__ __align__(16) float    row[D];     // 32 KB: whole row image
    __shared__ uint64_t               comp[NPK];  //  4 KB: composite sort keys

    const unsigned t = threadIdx.x;
    const size_t   rbase = (size_t)blockIdx.x * (size_t)D;

    // ---------- Phase 1: async global -> LDS copy of the row (32 KB) ----------
    // Each lane moves 4 x 16B chunks; chunk c = t + pass*T, byte offset c*16.
    {
        unsigned long long gbase = (unsigned long long)(const void*)(z + rbase);
#pragma unroll
        for (int pass = 0; pass < 4; ++pass) {
            unsigned c    = t + (unsigned)pass * T;            // 16B chunk id
            unsigned goff = c * 16u;                           // global byte off
            // LDS byte address = low 32 bits of the generic pointer (ISA 10.2)
            unsigned ldst = (unsigned)(unsigned long long)(const void*)&row[c * 4u];
            asm volatile("global_load_async_to_lds_b128 %0, %1, %2 offset:0 th:TH_LOAD_NT"
                         :: "v"(ldst), "v"(goff), "s"(gbase) : "memory");
        }
        // Each wave waits for its own async transfers, then the workgroup
        // barrier makes the whole row visible to everyone.
        asm volatile("s_wait_asynccnt 0" ::: "memory");
    }
    __syncthreads();

    // ---------- Phase 2: composite keys ----------
    // Map f32 to monotonic u32 (IEEE total order for non-NaN), append index
    // so all 512 keys are distinct -> any comparison sort == stable argsort.
    {
        unsigned ku = __float_as_uint(row[t * P]);
        ku = (ku & 0x80000000u) ? ~ku : (ku | 0x80000000u);
        comp[t] = ((uint64_t)ku << 32) | (uint64_t)t;
    }

    // ---------- Phase 3: bitonic sort of 512 composites in LDS ----------
    for (unsigned k = 2; k <= (unsigned)NPK; k <<= 1) {
        for (unsigned j = k >> 1; j > 0; j >>= 1) {
            __syncthreads();
            unsigned ixj = t ^ j;
            if (ixj > t) {                       // owner of the pair
                uint64_t a = comp[t];
                uint64_t b = comp[ixj];
                bool up = ((t & k) == 0);
                if ((a > b) == up) { comp[t] = b; comp[ixj] = a; }
            }
        }
    }
    __syncthreads();

    // ---------- Phase 4: gather from LDS, coalesced NT b128 stores ----------
    // Output chunk c (16B) belongs to output packet p=c/4, word w=c%4 and
    // sources input packet inds[p] = low 32 bits of sorted composite.
#pragma unroll
    for (int pass = 0; pass < 4; ++pass) {
        unsigned c = t + (unsigned)pass * T;          // 0..2047
        unsigned p = c >> 2;
        unsigned w = c & 3u;
        unsigned src = (unsigned)(comp[p] & 0xffffffffu);
        const v4f v = *(const v4f*)&row[src * P + w * 4u];        // ds_load_b128
        __builtin_nontemporal_store(v, (v4f*)(out + rbase + (size_t)c * 4u));
    }
}

extern "C" void kernel_launch(void* const* d_in, const int* in_sizes, int n_in,
                              void* d_out, int out_size, void* d_ws, size_t ws_size,
                              hipStream_t stream) {
    const float* z = (const float*)d_in[0];
    float* out     = (float*)d_out;
    const int B    = in_sizes[0] / D;   // 4096 rows for the reference shapes
    sort_latent_kernel<<<dim3(B), dim3(T), 0, stream>>>(z, out);
}